// GMM_21483426415100
// MI455X (gfx1250) — compile-verified
//
#include <hip/hip_runtime.h>
#include <hip/hip_bf16.h>
#include <math.h>

typedef __attribute__((ext_vector_type(2))) float v2f;
typedef __attribute__((ext_vector_type(4))) float v4f;
typedef __attribute__((ext_vector_type(8))) float v8f;

#define GMM_D 64
#define GMM_K 32
#define GMM_LOG2PI 1.8378770664093453f
#define GMM_ALPHA0 1000.0f
#define TILES_PER_BLOCK 8
#define NTHREADS 256

// ---------------------------------------------------------------------------
// Main kernel: one wave32 per 16-row tile of x.
// Fused GEMM  G[n,k] = sum_d x[n,d]^2 * iv[k,d]  +  x[n,d] * (-2*mu*iv)[k,d]
// via V_WMMA_F32_16X16X4_F32 (M=16,N=16,K=4), inner dim 2*D=128 -> 64 WMMA/tile.
// Then comp = c[k] - 0.5*G, per-row logsumexp over K=32 via lane butterflies.
// ---------------------------------------------------------------------------
__global__ __launch_bounds__(NTHREADS)
void gmm_main_kernel(const float* __restrict__ x,
                     const float* __restrict__ theta,
                     float* __restrict__ partial,
                     int nRows, int nTiles)
{
    // Pair-interleaved weights: sW[((t*32 + d/2)*32 + k)*2 + (d&1)]
    //   t=0 -> iv[k,d],  t=1 -> -2*mu[k,d]*iv[k,d]
    __shared__ float sW[2 * GMM_D * GMM_K];           // 16 KB
    __shared__ float sC[GMM_K];                       // per-component constant
    __shared__ float sX[TILES_PER_BLOCK * 16 * GMM_D]; // 32 KB x staging

    const int tid = threadIdx.x;

    // ---- Phase 1: per-component weights + constants (threads 0..31, k = tid)
    if (tid < GMM_K) {
        const int k = tid;
        const float* muk = theta + GMM_K + k * GMM_D;
        const float* sgk = theta + GMM_K + GMM_K * GMM_D + k * GMM_D;
        float logsig = 0.0f, m2 = 0.0f;
        for (int d = 0; d < GMM_D; ++d) {
            float mu = muk[d];
            float sg = sgk[d];
            float iv = 1.0f / (sg * sg);
            float w  = -2.0f * mu * iv;
            logsig += __logf(sg);
            m2     += mu * mu * iv;
            int dp = d >> 1, lh = d & 1;
            sW[(((0 * 32 + dp) * 32) + k) * 2 + lh] = iv;
            sW[(((1 * 32 + dp) * 32) + k) * 2 + lh] = w;
        }
        sC[k] = __logf(theta[k]) - logsig
              - 0.5f * (float)GMM_D * GMM_LOG2PI - 0.5f * m2;
    }

    // ---- Phase 1b: coalesced stage of 128 rows of x into LDS (b128 loads)
    {
        const long base  = (long)blockIdx.x * (TILES_PER_BLOCK * 16 * GMM_D);
        const long total = (long)nRows * GMM_D;
        #pragma unroll
        for (int i = 0; i < (TILES_PER_BLOCK * 16 * GMM_D) / (NTHREADS * 4); ++i) {
            long g = base + (long)(i * NTHREADS + tid) * 4;
            v4f v = {0.0f, 0.0f, 0.0f, 0.0f};
            if (g + 3 < total) v = *(const v4f*)(x + g);
            *(v4f*)(sX + (i * NTHREADS + tid) * 4) = v;
        }
    }
    __syncthreads();

    // ---- Phase 2: WMMA GEMM per wave
    const int wave = tid >> 5;
    const int lane = tid & 31;
    const int tile = blockIdx.x * TILES_PER_BLOCK + wave;
    if (tile >= nTiles) return;

    const int m   = lane & 15;   // A row / C column index
    const int hi  = lane >> 4;   // K-pair selector for A/B operands
    const int col = m;           // B column within a 16-wide K-half

    v8f acc0 = {0.f,0.f,0.f,0.f,0.f,0.f,0.f,0.f};
    v8f acc1 = {0.f,0.f,0.f,0.f,0.f,0.f,0.f,0.f};
    const float* xrow = sX + (wave * 16 + m) * GMM_D;

    #pragma unroll
    for (int j = 0; j < 16; ++j) {              // d0 = 4*j, chunk of 4
        v2f a  = *(const v2f*)(xrow + 4 * j + 2 * hi);
        v2f a2 = a * a;
        const int dp = 2 * j + hi;
        v2f b0h0 = *(const v2f*)(sW + (((0 * 32 + dp) * 32) + col) * 2);
        v2f b1h0 = *(const v2f*)(sW + (((1 * 32 + dp) * 32) + col) * 2);
        v2f b0h1 = *(const v2f*)(sW + (((0 * 32 + dp) * 32) + col + 16) * 2);
        v2f b1h1 = *(const v2f*)(sW + (((1 * 32 + dp) * 32) + col + 16) * 2);
        acc0 = __builtin_amdgcn_wmma_f32_16x16x4_f32(false, a2, false, b0h0,
                                                     (short)0, acc0, false, false);
        acc0 = __builtin_amdgcn_wmma_f32_16x16x4_f32(false, a,  false, b1h0,
                                                     (short)0, acc0, false, false);
        acc1 = __builtin_amdgcn_wmma_f32_16x16x4_f32(false, a2, false, b0h1,
                                                     (short)0, acc1, false, false);
        acc1 = __builtin_amdgcn_wmma_f32_16x16x4_f32(false, a,  false, b1h1,
                                                     (short)0, acc1, false, false);
    }

    // ---- Phase 3: comp = c[k] - 0.5*G ; per-row logsumexp over 32 components
    const float ck0 = sC[col];
    const float ck1 = sC[col + 16];
    float acc = 0.0f;
    #pragma unroll
    for (int i = 0; i < 8; ++i) {
        // C layout: VGPR i -> row (i) in lanes 0..15, row (i+8) in lanes 16..31
        float v0 = ck0 - 0.5f * acc0[i];
        float v1 = ck1 - 0.5f * acc1[i];
        float mx = fmaxf(v0, v1);
        mx = fmaxf(mx, __shfl_xor(mx, 1, 32));
        mx = fmaxf(mx, __shfl_xor(mx, 2, 32));
        mx = fmaxf(mx, __shfl_xor(mx, 4, 32));
        mx = fmaxf(mx, __shfl_xor(mx, 8, 32));
        float sm = __expf(v0 - mx) + __expf(v1 - mx);
        sm += __shfl_xor(sm, 1, 32);
        sm += __shfl_xor(sm, 2, 32);
        sm += __shfl_xor(sm, 4, 32);
        sm += __shfl_xor(sm, 8, 32);
        int rowIdx = tile * 16 + i + 8 * hi;
        if (rowIdx < nRows) acc += mx + __logf(sm);
    }
    acc += __shfl_xor(acc, 16, 32);   // combine rows 0..7 and 8..15 halves
    if (lane == 0) partial[tile] = acc;
}

// ---------------------------------------------------------------------------
// Deterministic finalize: fixed-order tree reduction of tile partials,
// priors over theta, scale by full_data_size / N. Single block.
// ---------------------------------------------------------------------------
__global__ __launch_bounds__(NTHREADS)
void gmm_finalize_kernel(const float* __restrict__ partial, int nTiles,
                         const float* __restrict__ theta,
                         const int* __restrict__ fullN,
                         float* __restrict__ out,
                         int nRows, float gammaConst)
{
    __shared__ float rLL[NTHREADS];
    __shared__ float rPR[NTHREADS];
    const int tid = threadIdx.x;

    float sll = 0.0f;
    for (int i = tid; i < nTiles; i += NTHREADS) sll += partial[i];

    float pr = 0.0f;
    for (int i = tid; i < GMM_K * GMM_D; i += NTHREADS) {
        float mu = theta[GMM_K + i];
        pr += -0.5f * mu * mu - 0.5f * GMM_LOG2PI;
        float sg = theta[GMM_K + GMM_K * GMM_D + i];
        float ls = __logf(sg);
        pr += -ls - 0.5f * GMM_LOG2PI - 0.5f * ls * ls;
    }
    if (tid < GMM_K) pr += (GMM_ALPHA0 - 1.0f) * __logf(theta[tid]);

    rLL[tid] = sll;
    rPR[tid] = pr;
    __syncthreads();
    for (int w = NTHREADS / 2; w > 0; w >>= 1) {
        if (tid < w) { rLL[tid] += rLL[tid + w]; rPR[tid] += rPR[tid + w]; }
        __syncthreads();
    }
    if (tid == 0) {
        float scale = (float)fullN[0] / (float)nRows;
        out[0] = rLL[0] * scale + rPR[0] + gammaConst;
    }
}

// ---------------------------------------------------------------------------
extern "C" void kernel_launch(void* const* d_in, const int* in_sizes, int n_in,
                              void* d_out, int out_size, void* d_ws, size_t ws_size,
                              hipStream_t stream)
{
    (void)n_in; (void)out_size; (void)ws_size;
    const float* x     = (const float*)d_in[0];
    const float* theta = (const float*)d_in[1];
    const int*   fullN = (const int*)d_in[2];

    const int nRows  = in_sizes[0] / GMM_D;        // 50000
    const int nTiles = (nRows + 15) / 16;          // 3125
    float* partial = (float*)d_ws;

    const int nBlocks = (nTiles + TILES_PER_BLOCK - 1) / TILES_PER_BLOCK;
    gmm_main_kernel<<<nBlocks, NTHREADS, 0, stream>>>(x, theta, partial,
                                                      nRows, nTiles);

    // gammaln(K*alpha0) - K*gammaln(alpha0): pure host-side constant
    const float gammaConst =
        (float)(lgamma((double)GMM_K * (double)GMM_ALPHA0)
                - (double)GMM_K * lgamma((double)GMM_ALPHA0));

    gmm_finalize_kernel<<<1, NTHREADS, 0, stream>>>(partial, nTiles, theta,
                                                    fullN, (float*)d_out,
                                                    nRows, gammaConst);
}